// GIN_54288386621898
// MI455X (gfx1250) — compile-verified
//
#include <hip/hip_runtime.h>

#define NN 40000
#define EE 640000
#define GG 256
#define HH 128
#define COUT 64
#define LLAYERS 5
#define BN_EPS 1e-5f
#define YSTRIDE 130   // padded LDS row stride (floats): conflict-free b64 reads

typedef float v2f __attribute__((ext_vector_type(2)));
typedef float v8f __attribute__((ext_vector_type(8)));

// ---------------- agg = h (self term of GIN) ----------------
__global__ void copy_f4(const float* __restrict__ src, float* __restrict__ dst, int n4) {
  int i = blockIdx.x * blockDim.x + threadIdx.x;
  if (i < n4) ((float4*)dst)[i] = ((const float4*)src)[i];
}

// ---------------- agg[dst] += h[src] over edges ----------------
__global__ void edge_scatter(const float* __restrict__ h,
                             const int* __restrict__ ei,   // [2, E]
                             float* __restrict__ agg) {
  int tid = blockIdx.x * blockDim.x + threadIdx.x;
  if (tid >= EE * 32) return;
  int e = tid >> 5;
  int q = (tid & 31) << 2;             // float4 chunk
  int s = ei[e];                        // src node
  int d = ei[EE + e];                   // dst node
  float4 v = *(const float4*)(h + (size_t)s * HH + q);
  float* p = agg + (size_t)d * HH + q;
  atomicAdd(p + 0, v.x);
  atomicAdd(p + 1, v.y);
  atomicAdd(p + 2, v.z);
  atomicAdd(p + 3, v.w);
}

// Stage W into LDS pre-swizzled so a B fragment is one aligned ds_load_b64:
//   Wswz[p*256 + c*2 + r] = W[(2p+r)*128 + c]   (p = K-pair, c = col, r = 0/1)
__device__ __forceinline__ void stage_w_swizzled(float* Wswz, const float* __restrict__ W,
                                                 int tid) {
  for (int i = tid; i < HH * HH / 2; i += 256) {   // i = p*128 + c
    int p = i >> 7;
    int c = i & 127;
    v2f w;
    w.x = W[(2 * p) * HH + c];
    w.y = W[(2 * p + 1) * HH + c];
    *(v2f*)(Wswz + p * 256 + c * 2) = w;
  }
}

// ---------------- fused  relu( relu(bn(x@W1+b1)) @ W2 + b2 ) ----------------
// 256 threads = 8 waves; each wave computes 16 rows x 128 cols with
// V_WMMA_F32_16X16X4_F32 (fp32 in/out, K=4 per instruction).
// One weight matrix resident at a time (~130 KB LDS -> 2 workgroups/WGP).
__global__ __launch_bounds__(256) void gin_mlp(
    const float* __restrict__ xin, const float* __restrict__ W1,
    const float* __restrict__ b1, const float* __restrict__ gamma,
    const float* __restrict__ beta, const float* __restrict__ rmean,
    const float* __restrict__ rvar, const float* __restrict__ W2,
    const float* __restrict__ b2, float* __restrict__ out) {
  extern __shared__ float smem[];
  float* Wswz = smem;                        // 64 * 256 = 16384 floats
  float* scl = Wswz + HH * HH;               // 128  (bn scale)
  float* bia = scl + HH;                     // 128  (fused bn bias incl. b1)
  float* b2s = bia + HH;                     // 128
  float* ystage = b2s + HH;                  // 8 * 16 * YSTRIDE

  int tid = threadIdx.x;
  stage_w_swizzled(Wswz, W1, tid);
  if (tid < HH) {
    float s = gamma[tid] * rsqrtf(rvar[tid] + BN_EPS);
    scl[tid] = s;
    bia[tid] = (b1[tid] - rmean[tid]) * s + beta[tid];
  } else if (tid < 2 * HH) {
    b2s[tid - HH] = b2[tid - HH];
  }
  __syncthreads();

  int wave = tid >> 5;
  int lane = tid & 31;
  int lr = lane & 15;                 // M index for A frags, N index for B/C frags
  int hi = lane >> 4;                 // upper half-wave handles K+2,K+3
  int tile = blockIdx.x * 8 + wave;   // 16-row tile id, 2500 total
  bool active = (tile * 16) < NN;
  float* ys = ystage + wave * 16 * YSTRIDE;

  v8f zero = {};
  v8f acc[8];

  // ---------------- GEMM1: x @ W1, fused BN + ReLU ----------------
  if (active) {
#pragma unroll
    for (int t = 0; t < 8; ++t) acc[t] = zero;
    const float* xr = xin + (size_t)(tile * 16 + lr) * HH + 2 * hi;
    for (int k = 0; k < HH; k += 4) {
      // A frag 16x4: lane<16 -> (M=lr, K=k,k+1); lane>=16 -> (M=lr, K=k+2,k+3)
      v2f a = *(const v2f*)(xr + k);
      const float* wrow = Wswz + ((k >> 1) + hi) * 256 + lr * 2;
#pragma unroll
      for (int t = 0; t < 8; ++t) {
        v2f b = *(const v2f*)(wrow + t * 32);   // single ds_load_b64
        acc[t] = __builtin_amdgcn_wmma_f32_16x16x4_f32(
            false, a, false, b, (short)0, acc[t], false, false);
      }
    }
    // BN + ReLU epilogue, restage (transposes N->K for GEMM2 via LDS)
#pragma unroll
    for (int t = 0; t < 8; ++t) {
      int col = t * 16 + lr;
      float s = scl[col], bb = bia[col];
#pragma unroll
      for (int v = 0; v < 8; ++v) {
        int m = (lane < 16) ? v : (v + 8);   // C layout: VGPR v -> M or M+8
        ys[m * YSTRIDE + col] = fmaxf(acc[t][v] * s + bb, 0.0f);
      }
    }
  }
  __syncthreads();                 // all waves done reading W1 / writing ystage
  stage_w_swizzled(Wswz, W2, tid); // overwrite with W2 (all threads participate)
  __syncthreads();

  // ---------------- GEMM2: y @ W2 + b2, ReLU ----------------
  if (active) {
#pragma unroll
    for (int t = 0; t < 8; ++t) acc[t] = zero;
    const float* yr = ys + lr * YSTRIDE + 2 * hi;
    for (int k = 0; k < HH; k += 4) {
      v2f a = *(const v2f*)(yr + k);          // ds_load_b64, conflict-free
      const float* wrow = Wswz + ((k >> 1) + hi) * 256 + lr * 2;
#pragma unroll
      for (int t = 0; t < 8; ++t) {
        v2f b = *(const v2f*)(wrow + t * 32);
        acc[t] = __builtin_amdgcn_wmma_f32_16x16x4_f32(
            false, a, false, b, (short)0, acc[t], false, false);
      }
    }
#pragma unroll
    for (int t = 0; t < 8; ++t) {
      int col = t * 16 + lr;
      float bb = b2s[col];
#pragma unroll
      for (int v = 0; v < 8; ++v) {
        int m = (lane < 16) ? v : (v + 8);
        out[(size_t)(tile * 16 + m) * HH + col] = fmaxf(acc[t][v] + bb, 0.0f);
      }
    }
  }
}

// ---------------- mean pool + head ----------------
__global__ void pool_zero(float* __restrict__ pooled, float* __restrict__ counts) {
  int i = blockIdx.x * blockDim.x + threadIdx.x;
  if (i < GG * HH) pooled[i] = 0.0f;
  if (i < GG) counts[i] = 0.0f;
}

__global__ void pool_scatter(const float* __restrict__ h, const int* __restrict__ batch,
                             float* __restrict__ pooled, float* __restrict__ counts) {
  int tid = blockIdx.x * blockDim.x + threadIdx.x;
  if (tid >= NN * 32) return;
  int n = tid >> 5;
  int q = (tid & 31) << 2;
  int g = batch[n];
  float4 v = *(const float4*)(h + (size_t)n * HH + q);
  float* p = pooled + (size_t)g * HH + q;
  atomicAdd(p + 0, v.x);
  atomicAdd(p + 1, v.y);
  atomicAdd(p + 2, v.z);
  atomicAdd(p + 3, v.w);
  if ((tid & 31) == 0) atomicAdd(counts + g, 1.0f);
}

__global__ void head_kernel(const float* __restrict__ pooled, const float* __restrict__ counts,
                            const float* __restrict__ l1w, const float* __restrict__ l1b,
                            const float* __restrict__ l2w, const float* __restrict__ l2b,
                            float* __restrict__ out) {
  __shared__ float p[HH];
  __shared__ float qv[HH];
  int g = blockIdx.x, j = threadIdx.x;
  float cnt = fmaxf(counts[g], 1.0f);
  p[j] = pooled[(size_t)g * HH + j] / cnt;
  __syncthreads();
  float acc = l1b[j];
  for (int k = 0; k < HH; ++k) acc = fmaf(p[k], l1w[k * HH + j], acc);
  qv[j] = fmaxf(acc, 0.0f);
  __syncthreads();
  if (j < COUT) {
    float a2 = l2b[j];
    for (int k = 0; k < HH; ++k) a2 = fmaf(qv[k], l2w[k * COUT + j], a2);
    out[(size_t)g * COUT + j] = a2;
  }
}

extern "C" void kernel_launch(void* const* d_in, const int* in_sizes, int n_in,
                              void* d_out, int out_size, void* d_ws, size_t ws_size,
                              hipStream_t stream) {
  const float* x     = (const float*)d_in[0];
  const int*   ei    = (const int*)d_in[1];
  const int*   batch = (const int*)d_in[2];
  const float* W1    = (const float*)d_in[3];
  const float* b1    = (const float*)d_in[4];
  const float* gamma = (const float*)d_in[5];
  const float* beta  = (const float*)d_in[6];
  const float* rmean = (const float*)d_in[7];
  const float* rvar  = (const float*)d_in[8];
  const float* W2    = (const float*)d_in[9];
  const float* b2    = (const float*)d_in[10];
  const float* l1w   = (const float*)d_in[11];
  const float* l1b   = (const float*)d_in[12];
  const float* l2w   = (const float*)d_in[13];
  const float* l2b   = (const float*)d_in[14];
  float* out = (float*)d_out;

  float* bufA   = (float*)d_ws;                 // N*H node features (layer out)
  float* bufB   = bufA + (size_t)NN * HH;       // N*H aggregation buffer
  float* pooled = bufB + (size_t)NN * HH;       // G*H
  float* counts = pooled + (size_t)GG * HH;     // G

  const int n4 = NN * HH / 4;
  const size_t mlp_lds = (size_t)(HH * HH + 3 * HH + 8 * 16 * YSTRIDE) * sizeof(float);
  const int mlp_blocks = (NN / 16 + 7) / 8;     // 313

  const float* hin = x;
  for (int l = 0; l < LLAYERS; ++l) {
    copy_f4<<<(n4 + 255) / 256, 256, 0, stream>>>(hin, bufB, n4);
    edge_scatter<<<(EE * 32 + 255) / 256, 256, 0, stream>>>(hin, ei, bufB);
    gin_mlp<<<mlp_blocks, 256, mlp_lds, stream>>>(
        bufB, W1 + (size_t)l * HH * HH, b1 + (size_t)l * HH,
        gamma + (size_t)l * HH, beta + (size_t)l * HH,
        rmean + (size_t)l * HH, rvar + (size_t)l * HH,
        W2 + (size_t)l * HH * HH, b2 + (size_t)l * HH, bufA);
    hin = bufA;
  }
  pool_zero<<<(GG * HH + 255) / 256, 256, 0, stream>>>(pooled, counts);
  pool_scatter<<<(NN * 32 + 255) / 256, 256, 0, stream>>>(bufA, batch, pooled, counts);
  head_kernel<<<GG, HH, 0, stream>>>(pooled, counts, l1w, l1b, l2w, l2b, out);
}